// SIFT_8332236554382
// MI455X (gfx1250) — compile-verified
//
#include <hip/hip_runtime.h>
#include <math.h>

// SIFT dense descriptor for 65x65 patches, one workgroup per (patch,channel) slice.
// - Patch staged to LDS via the Tensor Data Mover (tensor_load_to_lds + s_wait_tensorcnt)
//   when available; cooperative global loads otherwise.
// - Angular soft-binning + separable triangular pooling: <=4 ds_add_f32 LDS atomics per
//   pixel into colsum[32][65], then a (32x65)x(65x4) GEMM on V_WMMA_F32_16X16X4_F32.
// - Branch-free triangular weights: for integer t, 13-|t-12.5| <= -0.5 outside [0,26),
//   so fmaxf(0, .) replaces all range guards (no EXEC-mask churn in the WMMA loop).

#define PS       65
#define NAB      8
#define NSLICE   6144      // 32*3*64
#define DESC     128

typedef __attribute__((ext_vector_type(2))) float v2f;
typedef __attribute__((ext_vector_type(8))) float v8f;

#if defined(__gfx1250__) && __has_builtin(__builtin_amdgcn_tensor_load_to_lds)
#define HAVE_TDM 1
typedef unsigned int uv4 __attribute__((ext_vector_type(4)));
typedef int iv4 __attribute__((ext_vector_type(4)));
typedef int iv8 __attribute__((ext_vector_type(8)));
#else
#define HAVE_TDM 0
#endif

// Triangular pooling weight, branch-free. Column version folds the 1/169 2-D norm.
__device__ __forceinline__ float triw(int winstart, int pos) {
    return fmaxf(0.0f, 13.0f - fabsf((float)(pos - winstart) - 12.5f));
}

__global__ __launch_bounds__(256) void sift_desc_kernel(const float* __restrict__ img,
                                                        float* __restrict__ out)
{
    __shared__ float patch[PS * PS];        // 16900 B, packed (TDM writes contiguous rows)
    __shared__ float g1d[PS];               // normalized 1-D gaussian
    __shared__ float colsum[32 * PS];       // [a*4+jy][q]  (8320 B)
    __shared__ float ang[DESC];
    __shared__ float red[1];

    const int m   = blockIdx.x;             // slice id: ((b*3+c)*8+i)*8+j
    const int tid = threadIdx.x;

    const int j  = m & 7;
    const int i  = (m >> 3) & 7;
    const int bc = m >> 6;                  // b*3 + c
    const long base = ((long)bc * 520 + (long)i * 65) * 520 + (long)j * 65;

#if HAVE_TDM
    // ---- async DMA of the 65x65 f32 tile (row stride 520 elems) into LDS ----
    if (tid < 32) {                          // wave 0 issues; EXEC-independent DMA
        unsigned long long ga = (unsigned long long)(const void*)(img + base);
        unsigned int ldsOff   = (unsigned int)(unsigned long long)(void*)patch;
        uv4 g0;
        g0.x = 1u;                                           // count=1 (valid user D#)
        g0.y = ldsOff;                                       // lds_addr (bytes)
        g0.z = (unsigned int)ga;                             // global_addr[31:0]
        g0.w = (unsigned int)((ga >> 32) & 0x01FFFFFFu)      // global_addr[56:32]
             | (2u << 30);                                   // type = 2 ("image")
        iv8 g1;
        g1[0] = (int)(2u << 16);   // workgroup_mask=0, data_size=2 (4 bytes)
        g1[1] = (int)(65u << 16);  // tensor_dim0 = 65  (bits 79:48, lo16 in dw1[31:16])
        g1[2] = (int)(65u << 16);  // tensor_dim1 = 65  (bits 111:80, lo16 in dw2[31:16])
        g1[3] = (int)(65u << 16);  // tile_dim0   = 65  (bits 127:112)
        g1[4] = 65;                // tile_dim1   = 65  (bits 143:128), tile_dim2 = 0
        g1[5] = 520;               // tensor_dim0_stride = 520 (bits 207:160, lo32)
        g1[6] = 0;                 // stride hi, tensor_dim1_stride lo
        g1[7] = 0;
        iv4 gz = {0, 0, 0, 0};     // groups 2/3 unused for a 2-D tile
#if __clang_major__ >= 23
        iv8 gz8 = {0, 0, 0, 0, 0, 0, 0, 0};
        __builtin_amdgcn_tensor_load_to_lds(g0, g1, gz, gz, gz8, 0);
#else
        __builtin_amdgcn_tensor_load_to_lds(g0, g1, gz, gz, 0);
#endif
        __builtin_amdgcn_s_wait_tensorcnt(0);
    }
#endif

    // ---- 1-D gaussian (sigma = 65/sqrt(2)), normalized to sum 1 ----
    if (tid < PS) {
        const float sigma = 65.0f * 0.70710678118654752f;
        float x = (float)tid - 32.0f;
        g1d[tid] = __expf(-(x * x) / (2.0f * sigma * sigma));
    }
    for (int k = tid; k < 32 * PS; k += 256) colsum[k] = 0.0f;

#if !HAVE_TDM
    for (int p = tid; p < PS * PS; p += 256) {
        int r = p / PS, q = p - r * PS;
        patch[r * PS + q] = img[base + (long)r * 520 + q];
    }
#endif
    __syncthreads();
    if (tid == 0) {
        float s = 0.0f;
        for (int k = 0; k < PS; ++k) s += g1d[k];
        red[0] = 1.0f / s;
    }
    __syncthreads();
    if (tid < PS) g1d[tid] *= red[0];
    __syncthreads();

    // ---- per-pixel: gradients, orientation, soft-binned row pooling ----
    for (int p = tid; p < PS * PS; p += 256) {
        int r = p / PS, q = p - r * PS;
        int rm = (r > 0) ? r - 1 : 0,  rp = (r < PS - 1) ? r + 1 : PS - 1;
        int qm = (q > 0) ? q - 1 : 0,  qp = (q < PS - 1) ? q + 1 : PS - 1;

        float gx = 0.5f * (patch[r  * PS + qp] - patch[r  * PS + qm]);
        float gy = 0.5f * (patch[rp * PS + q ] - patch[rm * PS + q ]);

        float mag = sqrtf(gx * gx + gy * gy + 1e-10f) * g1d[r] * g1d[q];
        float ori = atan2f(gy, gx + 1e-10f) + 6.28318530717958648f;

        float ob   = ori * 1.27323954473516269f;   // * NAB / (2*pi)
        float bf   = floorf(ob);
        float frac = ob - bf;
        float w0 = (1.0f - frac) * mag;
        float w1 = frac * mag;
        int b0 = ((int)bf) & (NAB - 1);
        int b1 = (b0 + 1) & (NAB - 1);

        #pragma unroll
        for (int jy = 0; jy < 4; ++jy) {
            float wr = triw(jy * 16 - 6, r);       // branch-free triangular weight
            if (wr > 0.0f) {                       // skip useless atomics only
                atomicAdd(&colsum[(b0 * 4 + jy) * PS + q], w0 * wr);
                atomicAdd(&colsum[(b1 * 4 + jy) * PS + q], w1 * wr);
            }
        }
    }
    __syncthreads();

    // ---- column pooling: ang(32x4) = colsum(32x65) * Wc(65x4) ----
#if __has_builtin(__builtin_amdgcn_wmma_f32_16x16x4_f32)
    {
        const int wave = tid >> 5;
        const int lane = tid & 31;
        if (wave < 2) {                      // 2 M-tiles of 16 rows; EXEC all-1 per wave
            const int tileM = wave * 16;
            const int lhi = lane >> 4;       // 0/1
            const int llo = lane & 15;
            const int rowA = tileM + llo;    // M index (rho = a*4+jy)
            const int ws   = llo * 16 - 6;   // window start for column bin N=llo
            v8f acc = {};
            #pragma unroll
            for (int kb = 0; kb < 17; ++kb) {
                int q0 = kb * 4 + 2 * lhi;   // K for VGPR 0
                int q1 = q0 + 1;             // K for VGPR 1
                v2f A, B;
                A.x = (q0 < PS) ? colsum[rowA * PS + q0] : 0.0f;
                A.y = (q1 < PS) ? colsum[rowA * PS + q1] : 0.0f;
                // Branch-free Wc: N>=4 / q>=65 columns are either discarded at the
                // D write or multiplied by a zero A element, so no guards needed.
                B.x = triw(ws, q0) * (1.0f / 169.0f);
                B.y = triw(ws, q1) * (1.0f / 169.0f);
                acc = __builtin_amdgcn_wmma_f32_16x16x4_f32(
                        false, A, false, B, (short)0, acc, false, false);
            }
            int N = llo;                     // output column (kx), valid < 4
            if (N < 4) {
                #pragma unroll
                for (int v = 0; v < 8; ++v) {
                    int rho = tileM + v + 8 * lhi;   // D layout: M = v + 8*(lane/16)
                    ang[rho * 4 + N] = acc[v];       // desc idx = a*16+jy*4+kx = 4*rho+kx
                }
            }
        }
    }
#else
    if (tid < DESC) {                        // scalar fallback: 128 dot products
        int rho = tid >> 2, kx = tid & 3;
        float s = 0.0f;
        for (int q = 0; q < PS; ++q)
            s += colsum[rho * PS + q] * triw(kx * 16 - 6, q) * (1.0f / 169.0f);
        ang[tid] = s;
    }
#endif
    __syncthreads();

    // ---- L2 norm -> clip(0.2) -> L2 norm ----
    if (tid == 0) {
        float s = 0.0f;
        for (int k = 0; k < DESC; ++k) s += ang[k] * ang[k];
        red[0] = sqrtf(s);
    }
    __syncthreads();
    float n1 = fmaxf(red[0], 1e-12f);
    if (tid < DESC) {
        float v = ang[tid] / n1;
        ang[tid] = fminf(fmaxf(v, 0.0f), 0.2f);
    }
    __syncthreads();
    if (tid == 0) {
        float s = 0.0f;
        for (int k = 0; k < DESC; ++k) s += ang[k] * ang[k];
        red[0] = sqrtf(s);
    }
    __syncthreads();
    float n2 = fmaxf(red[0], 1e-12f);
    if (tid < DESC) out[(long)m * DESC + tid] = ang[tid] / n2;
}

extern "C" void kernel_launch(void* const* d_in, const int* in_sizes, int n_in,
                              void* d_out, int out_size, void* d_ws, size_t ws_size,
                              hipStream_t stream) {
    (void)in_sizes; (void)n_in; (void)out_size; (void)d_ws; (void)ws_size;
    const float* img = (const float*)d_in[0];
    float* out = (float*)d_out;
    sift_desc_kernel<<<NSLICE, 256, 0, stream>>>(img, out);
}